// SimpleGraphSAGE_31344671326737
// MI455X (gfx1250) — compile-verified
//
#include <hip/hip_runtime.h>
#include <hip/hip_bf16.h>

#define N_NODES 100000
#define N_EDGES 1000000
#define IN_DIM  64
#define HID_DIM 64
#define OUT_DIM 2
#define KCAT    128              // [mean | x] concatenated K dimension
#define M_TILES (N_NODES / 16)   // 6250, exact

typedef __attribute__((ext_vector_type(2))) float v2f;
typedef __attribute__((ext_vector_type(8))) float v8f;

// ---------------------------------------------------------------------------
// Degree: one f32 atomic per edge.
__global__ void __launch_bounds__(256)
degree_kernel(const long long* __restrict__ dst, float* __restrict__ deg, int n_edges) {
    int e = blockIdx.x * blockDim.x + threadIdx.x;
    if (e < n_edges) atomicAdd(&deg[(int)dst[e]], 1.0f);
}

__global__ void __launch_bounds__(256)
invdeg_kernel(const float* __restrict__ deg, float* __restrict__ invdeg, int n) {
    int i = blockIdx.x * blockDim.x + threadIdx.x;
    if (i < n) invdeg[i] = 1.0f / fmaxf(deg[i], 1.0f);
}

// ---------------------------------------------------------------------------
// Scatter-add of 64-dim rows: thread = (edge, 4-dim chunk). 16 threads per
// edge -> one b128 gather of the source row chunk + 4 f32 atomics into the
// destination row. Lanes 0-15 of a wave cover one edge's full row (256 B,
// coalesced); f32 atomics are scalar on CDNA5 so 64/edge is the floor.
__global__ void __launch_bounds__(256)
scatter_sum_kernel(const long long* __restrict__ src, const long long* __restrict__ dst,
                   const float* __restrict__ feat, float* __restrict__ sum, int n_edges) {
    long long tid = (long long)blockIdx.x * blockDim.x + threadIdx.x;
    int e  = (int)(tid >> 4);
    int d4 = (int)(tid & 15) << 2;         // dim offset: 0,4,...,60
    if (e >= n_edges) return;
    int s = (int)src[e];
    int t = (int)dst[e];
    const float4 v = *(const float4*)(feat + (long long)s * 64 + d4);
    float* __restrict__ srow = sum + (long long)t * 64 + d4;
    atomicAdd(srow + 0, v.x);
    atomicAdd(srow + 1, v.y);
    atomicAdd(srow + 2, v.z);
    atomicAdd(srow + 3, v.w);
}

// ---------------------------------------------------------------------------
// Layer 1: h = relu([mean | x] @ [Wl1^T ; Wr1^T] + b1), all f32, via
// V_WMMA_F32_16X16X4_F32. One wave computes a 16x64 output tile.
//   A frag (16x4 f32, 2 VGPR): lanes 0-15 hold K=k0,k0+1 ; lanes 16-31 K=k0+2,k0+3
//   B frag (4x16 f32, 2 VGPR): mirrored, columns across lanes
//   C/D  (16x16 f32, 8 VGPR): VGPR v -> M=v (lanes 0-15) / M=v+8 (lanes 16-31)
// LDS holds the combined weight matrix K-pair-interleaved so each B fragment
// is one aligned ds_load_b64 straight into an even VGPR pair (no re-packing):
//   Bs[k/2][n] = { B[k][n], B[k+1][n] },  B = [Wl1^T ; Wr1^T]  (128 x 64)
__global__ void __launch_bounds__(256)
gemm1_wmma_kernel(const float* __restrict__ x, const float* __restrict__ sum1,
                  const float* __restrict__ invdeg,
                  const float* __restrict__ Wl1, const float* __restrict__ Wr1,
                  const float* __restrict__ b1, float* __restrict__ h) {
    __shared__ float2 Bs[(KCAT / 2) * HID_DIM];   // 64 x 64 float2 = 32 KB

    // Stage: i -> (n, k2); B[k][n] = Wl1[n][k] (k<64) / Wr1[n][k-64]
    for (int i = threadIdx.x; i < HID_DIM * (IN_DIM / 2); i += 256) {
        int n  = i >> 5;           // 0..63
        int k2 = i & 31;           // K pair index within one half
        const float* wl = Wl1 + n * IN_DIM + 2 * k2;
        const float* wr = Wr1 + n * IN_DIM + 2 * k2;
        Bs[k2 * HID_DIM + n]        = make_float2(wl[0], wl[1]);
        Bs[(k2 + 32) * HID_DIM + n] = make_float2(wr[0], wr[1]);
    }
    __syncthreads();

    int wave = threadIdx.x >> 5;
    int lane = threadIdx.x & 31;
    int tile = blockIdx.x * 8 + wave;
    if (tile >= M_TILES) return;           // wave-uniform: EXEC stays all-1s

    int rowbase = tile * 16;
    int m       = lane & 15;
    int khalf   = lane >> 4;               // 0 -> K pair (k0,k0+1), 1 -> (k0+2,k0+3)
    int row     = rowbase + m;
    float idg   = invdeg[row];
    const float* srow = sum1 + (long long)row * 64;
    const float* xrow = x    + (long long)row * 64;

    v8f acc[4];
    #pragma unroll
    for (int nt = 0; nt < 4; ++nt) {       // bias folded into C
        float bv = b1[nt * 16 + m];
        acc[nt] = (v8f){bv, bv, bv, bv, bv, bv, bv, bv};
    }

    const int klo = khalf << 1;            // lane's K sub-offset within each step

    // --- mean half: A[m][k] = sum1[m][k] * invdeg[m], k = 0..63 ---
    #pragma unroll 4
    for (int k0 = 0; k0 < 64; k0 += 4) {
        int kk = k0 + klo;
        v2f a;
        a[0] = srow[kk]     * idg;
        a[1] = srow[kk + 1] * idg;
        #pragma unroll
        for (int nt = 0; nt < 4; ++nt) {
            float2 bp = Bs[(kk >> 1) * HID_DIM + nt * 16 + m];
            v2f b = {bp.x, bp.y};
            acc[nt] = __builtin_amdgcn_wmma_f32_16x16x4_f32(
                false, a, false, b, (short)0, acc[nt], false, false);
        }
    }

    // --- self half: A[m][64+k] = x[m][k], k = 0..63 ---
    #pragma unroll 4
    for (int k0 = 0; k0 < 64; k0 += 4) {
        int kk = k0 + klo;
        v2f a;
        a[0] = xrow[kk];
        a[1] = xrow[kk + 1];
        #pragma unroll
        for (int nt = 0; nt < 4; ++nt) {
            float2 bp = Bs[((kk >> 1) + 32) * HID_DIM + nt * 16 + m];
            v2f b = {bp.x, bp.y};
            acc[nt] = __builtin_amdgcn_wmma_f32_16x16x4_f32(
                false, a, false, b, (short)0, acc[nt], false, false);
        }
    }

    // ReLU + store: D VGPR v -> row rowbase + v + 8*khalf, col nt*16 + m
    #pragma unroll
    for (int nt = 0; nt < 4; ++nt) {
        int c = nt * 16 + m;
        #pragma unroll
        for (int v = 0; v < 8; ++v) {
            int r = rowbase + v + (khalf << 3);
            h[(long long)r * 64 + c] = fmaxf(acc[nt][v], 0.0f);
        }
    }
}

// ---------------------------------------------------------------------------
// Layer 2: out = mean2 @ Wl2^T + h @ Wr2^T + b2 (N=2 only -> VALU + shuffle).
// One wave per node; lane covers dims lane and lane+32.
__global__ void __launch_bounds__(256)
layer2_kernel(const float* __restrict__ h, const float* __restrict__ sum2,
              const float* __restrict__ invdeg,
              const float* __restrict__ Wl2, const float* __restrict__ Wr2,
              const float* __restrict__ b2, float* __restrict__ out, int n_nodes) {
    int wave = threadIdx.x >> 5;
    int lane = threadIdx.x & 31;
    int n = blockIdx.x * 8 + wave;
    if (n >= n_nodes) return;

    float idg = invdeg[n];
    const float* hrow = h    + (long long)n * 64;
    const float* srow = sum2 + (long long)n * 64;

    float s0 = 0.0f, s1 = 0.0f;
    #pragma unroll
    for (int j = 0; j < 2; ++j) {
        int d    = lane + j * 32;
        float mv = srow[d] * idg;
        float hv = hrow[d];
        s0 += mv * Wl2[d]      + hv * Wr2[d];
        s1 += mv * Wl2[64 + d] + hv * Wr2[64 + d];
    }
    for (int off = 16; off > 0; off >>= 1) {
        s0 += __shfl_down(s0, off, 32);
        s1 += __shfl_down(s1, off, 32);
    }
    if (lane == 0) {
        out[(long long)n * 2 + 0] = s0 + b2[0];
        out[(long long)n * 2 + 1] = s1 + b2[1];
    }
}

// ---------------------------------------------------------------------------
extern "C" void kernel_launch(void* const* d_in, const int* in_sizes, int n_in,
                              void* d_out, int out_size, void* d_ws, size_t ws_size,
                              hipStream_t stream) {
    const float*     x   = (const float*)d_in[0];
    const long long* ei  = (const long long*)d_in[1];   // int64 per reference
    const long long* src = ei;
    const long long* dst = ei + N_EDGES;
    const float* Wl1 = (const float*)d_in[2];
    const float* Wr1 = (const float*)d_in[3];
    const float* b1  = (const float*)d_in[4];
    const float* Wl2 = (const float*)d_in[5];
    const float* Wr2 = (const float*)d_in[6];
    const float* b2  = (const float*)d_in[7];
    float* out = (float*)d_out;

    // Workspace layout (floats). deg/sum1/sum2 kept contiguous -> one memset.
    const size_t N_PAD = 100096;                 // N_NODES padded to 64
    const size_t ROWF  = (size_t)N_NODES * 64;   // 6,400,000
    float* deg    = (float*)d_ws;
    float* sum1   = deg + N_PAD;
    float* sum2   = sum1 + ROWF;
    float* invdeg = sum2 + ROWF;
    float* hbuf   = invdeg + N_PAD;

    // Zero accumulators (must be per-call: sums are accumulated with atomics).
    hipMemsetAsync(deg, 0, (N_PAD + 2 * ROWF) * sizeof(float), stream);

    // Degrees and reciprocal (shared by both layers).
    degree_kernel<<<(N_EDGES + 255) / 256, 256, 0, stream>>>(dst, deg, N_EDGES);
    invdeg_kernel<<<(N_NODES + 255) / 256, 256, 0, stream>>>(deg, invdeg, N_NODES);

    // Layer 1: aggregate x, then fused WMMA GEMM + bias + ReLU.
    {
        long long t = (long long)N_EDGES * 16;   // 16 threads per edge
        scatter_sum_kernel<<<(unsigned)((t + 255) / 256), 256, 0, stream>>>(
            src, dst, x, sum1, N_EDGES);
    }
    gemm1_wmma_kernel<<<(M_TILES + 7) / 8, 256, 0, stream>>>(
        x, sum1, invdeg, Wl1, Wr1, b1, hbuf);

    // Layer 2: aggregate h, then skinny projection to 2 outputs.
    {
        long long t = (long long)N_EDGES * 16;
        scatter_sum_kernel<<<(unsigned)((t + 255) / 256), 256, 0, stream>>>(
            src, dst, hbuf, sum2, N_EDGES);
    }
    layer2_kernel<<<(N_NODES + 7) / 8, 256, 0, stream>>>(
        hbuf, sum2, invdeg, Wl2, Wr2, b2, out, N_NODES);
}